// Mask2CubeManual_81965155877451
// MI455X (gfx1250) — compile-verified
//
#include <hip/hip_runtime.h>
#include <stdint.h>

#define Hh 256
#define HH (Hh*Hh)
#define TOPK 200
#define MIN_COUNT 400
#define FOCALF 221.0f

__device__ __forceinline__ float waveRedF(float v){
#pragma unroll
  for (int o = 16; o >= 1; o >>= 1) v += __shfl_xor(v, o, 32);
  return v;
}
__device__ __forceinline__ int waveRedI(int v){
#pragma unroll
  for (int o = 16; o >= 1; o >>= 1) v += __shfl_xor(v, o, 32);
  return v;
}

__global__ __launch_bounds__(256)
void mask2cube_kernel(const float* __restrict__ x, float* __restrict__ out){
  // 4-deep ring of (8 rows * 256 cols) slabs for async DMA staging (32KB)
  __shared__ float tile[4][8 * Hh];
  __shared__ float   wsumX[Hh];  __shared__ unsigned cntX[Hh];
  __shared__ float   wsumY[Hh];  __shared__ unsigned cntY[Hh];
  __shared__ unsigned totCnt;
  __shared__ int   dirC[4], dirR[4];
  __shared__ float dirNum[4], dirDen[4], pden[4];

  const int t = threadIdx.x;
  const int w = t >> 5, l = t & 31;
  const int b = blockIdx.x;
  const float* base = x + (size_t)b * HH;

  wsumX[t] = 0.f; cntX[t] = 0u;
  if (t == 0) totCnt = 0u;

  // fixed column set per lane: cols {4l..4l+3, 128+4l..128+4l+3}
  float    colW[8];
  unsigned colC[8];
#pragma unroll
  for (int j = 0; j < 8; ++j){ colW[j] = 0.f; colC[j] = 0u; }

  // ---------------- async DMA pipeline: 32 slabs of 8 rows (8KB), ring of 4 ----------------
  const uint64_t gbase   = (uint64_t)(uintptr_t)base + (uint64_t)t * 16u;
  const unsigned ldsBase = (unsigned)(uintptr_t)(&tile[0][0]) + (unsigned)t * 16u;

  // prologue: slabs 0..2 -> buffers 0..2 (each lane moves 2 x 16B per slab)
#pragma unroll
  for (int k = 0; k < 3; ++k){
    uint64_t ga = gbase + (uint64_t)k * 8192u;
    unsigned la = ldsBase + (unsigned)k * 8192u;
    asm volatile("global_load_async_to_lds_b128 %0, %1, off\n\t"
                 "global_load_async_to_lds_b128 %0, %1, off offset:4096"
                 :: "v"(la), "v"(ga) : "memory");
  }
  asm volatile("s_wait_asynccnt 0x4" ::: "memory");   // slab 0 DMA (this wave) complete

  for (int s = 0; s < 32; ++s){
    // publishes slab s (guaranteed done by previous iteration's wait) to all waves,
    // and guarantees buffer (s+3)&3 (consumed at iteration s-1) is free for reuse.
    __syncthreads();

    if (s + 3 < 32){
      uint64_t ga = gbase + (uint64_t)(s + 3) * 8192u;
      unsigned la = ldsBase + (unsigned)((s + 3) & 3) * 8192u;
      asm volatile("global_load_async_to_lds_b128 %0, %1, off\n\t"
                   "global_load_async_to_lds_b128 %0, %1, off offset:4096"
                   :: "v"(la), "v"(ga) : "memory");
    }
    // ensure this wave's slab s+1 DMA is complete (in-order completion):
    if (s <= 28)      asm volatile("s_wait_asynccnt 0x4" ::: "memory");
    else if (s == 29) asm volatile("s_wait_asynccnt 0x2" ::: "memory");
    else if (s == 30) asm volatile("s_wait_asynccnt 0x0" ::: "memory");

    // wave w processes row (s*8 + w); lane l handles 8 fixed columns
    const float* row = &tile[s & 3][w * Hh];
    float4 v0 = *(const float4*)(row + 4 * l);
    float4 v1 = *(const float4*)(row + 128 + 4 * l);
    float vv[8] = {v0.x, v0.y, v0.z, v0.w, v1.x, v1.y, v1.z, v1.w};

    float rowW = 0.f; int rowC = 0;
#pragma unroll
    for (int j = 0; j < 8; ++j){
      bool  m  = vv[j] > 0.5f;
      float mv = m ? vv[j] : 0.f;
      colW[j] += mv;
      colC[j] += m ? 1u : 0u;
      rowW    += mv;
      rowC    += m ? 1 : 0;
    }
    rowW = waveRedF(rowW);
    rowC = waveRedI(rowC);
    if (l == 0){
      int yy = s * 8 + w;
      wsumY[yy] = rowW;
      cntY[yy]  = (unsigned)rowC;
    }
  }
  __syncthreads();

  // fold per-lane column accumulators (8 waves contend per column)
#pragma unroll
  for (int j = 0; j < 8; ++j){
    int c = (j < 4) ? (4 * l + j) : (128 + 4 * l + (j - 4));
    atomicAdd(&wsumX[c], colW[j]);
    atomicAdd(&cntX[c],  colC[j]);
  }
  // total masked count
  {
    int rc = waveRedI((int)cntY[t]);
    if (l == 0) atomicAdd(&totCnt, (unsigned)rc);
  }
  __syncthreads();

  // ---------------- cutoff scan: dirs 0=x_max 1=x_min 2=y_max 3=y_min ----------------
  if (t < 4){
    const bool isX  = (t < 2);
    const bool isMx = ((t & 1) == 0);
    const unsigned* cn = isX ? cntX  : cntY;
    const float*    ws = isX ? wsumX : wsumY;
    int acc = 0, cstar = 0, r = 0;
    float num = 0.f, den = 0.f;
    for (int i = 0; i < Hh; ++i){
      int c = isMx ? (Hh - 1 - i) : i;
      int k = (int)cn[c];
      if (acc + k >= TOPK){ cstar = c; r = TOPK - acc; break; }
      acc += k;
      num += (float)c * ws[c];
      den += ws[c];
    }
    dirC[t] = cstar; dirR[t] = r; dirNum[t] = num; dirDen[t] = den;
  }
  __syncthreads();

  // ---------------- ordered partial sum at cutoff line (stable tie-break) ----------------
  if (w < 4){
    const int cstar = dirC[w], r = dirR[w];
    float myden = 0.f;
    int accc = 0;
    for (int ch = 0; ch < 8 && accc < r; ++ch){
      int idx = ch * 32 + l;
      size_t off = (w < 2) ? ((size_t)idx * Hh + cstar)    // column line: ascending y
                           : ((size_t)cstar * Hh + idx);   // row line: ascending x
      float v = base[off];
      bool  m = v > 0.5f;
      unsigned bal = (unsigned)__ballot(m);
      int before = accc + __popc(bal & ((1u << l) - 1u));
      float tw = (m && before < r) ? v : 0.f;
      myden += waveRedF(tw);
      accc  += __popc(bal);
    }
    if (l == 0) pden[w] = myden;
  }
  __syncthreads();

  // ---------------- geometry + output ----------------
  if (t == 0){
    float* o = out + (size_t)b * 7;
    if (totCnt > MIN_COUNT){
      float vals[4];
#pragma unroll
      for (int d = 0; d < 4; ++d){
        float pd = pden[d];
        vals[d] = (dirNum[d] + (float)dirC[d] * pd) / (dirDen[d] + pd);
      }
      float x_max = vals[0], x_min = vals[1], y_max = vals[2], y_min = vals[3];
      float ny_min = 255.f - y_max;
      float ny_max = 255.f - y_min;
      float z  = 1.f + ny_min / 128.f;
      float xm = x_min - 128.f, xM = x_max - 128.f;
      float inv = 1.f / (FOCALF * z);
      float x3_min = xm * inv,     x3_max = xM * inv;
      float y3_min = ny_min * inv, y3_max = ny_max * inv;
      o[0] = (x3_max + x3_min) * 0.5f;
      o[1] = (y3_max + y3_min) * 0.5f;
      o[2] = z;
      o[3] = (x3_max - x3_min) * 0.5f;
      o[4] = (y3_max - y3_min) * 0.5f;
      o[5] = 0.1f;
      o[6] = 0.f;
    } else {
#pragma unroll
      for (int i = 0; i < 7; ++i) o[i] = 0.f;
    }
  }
}

extern "C" void kernel_launch(void* const* d_in, const int* in_sizes, int n_in,
                              void* d_out, int out_size, void* d_ws, size_t ws_size,
                              hipStream_t stream){
  (void)n_in; (void)d_ws; (void)ws_size; (void)out_size;
  const float* x = (const float*)d_in[0];
  float* out = (float*)d_out;
  const int B = in_sizes[0] / HH;
  mask2cube_kernel<<<dim3(B), dim3(256), 0, stream>>>(x, out);
}